// Arap_42880953484257
// MI455X (gfx1250) — compile-verified
//
#include <hip/hip_runtime.h>
#include <stdint.h>

#define B_   2
#define N_   8192
#define K_   16
#define TPB  256
#define TILES_PER_BATCH (N_ / TPB)  // 32

typedef uint32_t u32x4 __attribute__((ext_vector_type(4)));
typedef uint32_t u32x8 __attribute__((ext_vector_type(8)));

// LDS byte offset of a (generic) pointer into __shared__ memory.
__device__ __forceinline__ uint32_t lds_addr_of(const void* p) {
  return (uint32_t)(uint64_t)(const __attribute__((address_space(3))) char*)p;
}

// One TDM descriptor: contiguous 1D copy of `nelem` f32 (nelem < 65536) global -> LDS.
// D# layout per CDNA5 ISA §8.3/§8.4 (group0 128b, group1 256b; groups 2/3 omitted => up-to-2D).
__device__ __forceinline__ void tdm_load_1d_f32(const float* gsrc, const float* ldst,
                                                uint32_t nelem) {
  const uint64_t ga = (uint64_t)(uintptr_t)gsrc;
  u32x4 g0;
  g0.x = 1u;                                                  // count=1, user descriptor
  g0.y = lds_addr_of(ldst);                                   // lds_addr (bytes)
  g0.z = (uint32_t)ga;                                        // global_addr[31:0]
  g0.w = (uint32_t)((ga >> 32) & 0x01FFFFFFu) | (2u << 30);   // global_addr[56:32], type=2
  u32x8 g1;
  g1.s0 = (2u << 16);                                         // data_size=2 (4 bytes); no mask/pad
  g1.s1 = (nelem & 0xFFFFu) << 16;                            // tensor_dim0[15:0] @ bits 63:48
  g1.s2 = (nelem >> 16) | (1u << 16);                         // tensor_dim0[31:16], tensor_dim1=1
  g1.s3 = (nelem & 0xFFFFu) << 16;                            // tile_dim0 @ bits 127:112
  g1.s4 = 0u;                                                 // tile_dim1=0, tile_dim2=0 (unused)
  g1.s5 = nelem;                                              // tensor_dim0_stride[31:0]
  g1.s6 = 0u;
  g1.s7 = 0u;
  asm volatile("tensor_load_to_lds %0, %1" :: "s"(g0), "s"(g1) : "memory");
}

// One cyclic Jacobi rotation zeroing A[P][Q] of symmetric A; accumulates V <- V*J.
template <int P, int Q, int RR>
__device__ __forceinline__ void jrot(float (&A)[3][3], float (&V)[3][3]) {
  const float apq = A[P][Q];
  float c = 1.0f, s = 0.0f;
  if (fabsf(apq) > 1e-30f) {
    const float tau = (A[Q][Q] - A[P][P]) / (2.0f * apq);
    const float t = copysignf(1.0f, tau) / (fabsf(tau) + sqrtf(1.0f + tau * tau));
    c = rsqrtf(1.0f + t * t);
    s = t * c;
  }
  const float app = A[P][P], aqq = A[Q][Q];
  A[P][P] = c * c * app - 2.0f * s * c * apq + s * s * aqq;
  A[Q][Q] = s * s * app + 2.0f * s * c * apq + c * c * aqq;
  A[P][Q] = 0.0f;
  A[Q][P] = 0.0f;
  const float arp = A[RR][P], arq = A[RR][Q];
  A[RR][P] = c * arp - s * arq;  A[P][RR] = A[RR][P];
  A[RR][Q] = s * arp + c * arq;  A[Q][RR] = A[RR][Q];
#pragma unroll
  for (int r = 0; r < 3; ++r) {
    const float vp = V[r][P], vq = V[r][Q];
    V[r][P] = c * vp - s * vq;
    V[r][Q] = s * vp + c * vq;
  }
}

__global__ __launch_bounds__(TPB, 1) void arap_kernel(
    const float* __restrict__ xyz1, const float* __restrict__ xyz2,
    const int* __restrict__ neighborList, const int* __restrict__ numNeighbors,
    const float* __restrict__ weightMatrix, float* __restrict__ out) {
  extern __shared__ float smem[];          // 192 KB: [0,3N) = xyz1[b], [3N,6N) = xyz2[b]
  float* s1 = smem;
  float* s2 = smem + 3 * N_;

  const int b = blockIdx.x / TILES_PER_BATCH;
  const int i = (blockIdx.x % TILES_PER_BATCH) * TPB + threadIdx.x;

  // --- Kick off the Tensor Data Mover DMA of this batch's coordinate arrays into LDS.
  // Wave 0 issues both descriptors (TDM ignores EXEC; other waves branch around).
  // Do NOT wait yet: TENSORcnt is independent of LOADcnt, so all waves can issue
  // their long-latency global gathers below while the 192 KB DMA is in flight.
  if (threadIdx.x < 32) {
    tdm_load_1d_f32(xyz1 + (size_t)b * 3 * N_, s1, 3 * N_);
    tdm_load_1d_f32(xyz2 + (size_t)b * 3 * N_, s2, 3 * N_);
  }

  const long long bn = (long long)b * N_ + i;
  const int*   nl = neighborList + bn * K_;
  const float* wr = weightMatrix + bn * (long long)N_;
  const int    nn = numNeighbors[bn];

  // Preload neighbor indices (coalesced b128s) and masked weights (random L2
  // gathers) -- overlapped with the TDM transfer.
  int jv[K_];
#pragma unroll
  for (int k = 0; k < K_; ++k) jv[k] = nl[k];
  float wv[K_];
#pragma unroll
  for (int k = 0; k < K_; ++k) wv[k] = (k < nn) ? wr[jv[k]] : 0.0f;

  // --- Now drain the DMA and make LDS visible to all waves.
  if (threadIdx.x < 32) {
    __builtin_amdgcn_s_wait_tensorcnt((short)0);
  }
  __syncthreads();

  const float cx1 = s1[3 * i + 0], cy1 = s1[3 * i + 1], cz1 = s1[3 * i + 2];
  const float cx2 = s2[3 * i + 0], cy2 = s2[3 * i + 1], cz2 = s2[3 * i + 2];

  // --- Pass 1: weighted covariance S[a][c] = sum_k w_k e1[a] e2[c]  (LDS gathers).
  float S[3][3] = {{0.0f, 0.0f, 0.0f}, {0.0f, 0.0f, 0.0f}, {0.0f, 0.0f, 0.0f}};
#pragma unroll
  for (int k = 0; k < K_; ++k) {
    const int j = jv[k];
    const float w = wv[k];
    const float e1x = cx1 - s1[3 * j + 0];
    const float e1y = cy1 - s1[3 * j + 1];
    const float e1z = cz1 - s1[3 * j + 2];
    const float e2x = cx2 - s2[3 * j + 0];
    const float e2y = cy2 - s2[3 * j + 1];
    const float e2z = cz2 - s2[3 * j + 2];
    S[0][0] += w * e1x * e2x; S[0][1] += w * e1x * e2y; S[0][2] += w * e1x * e2z;
    S[1][0] += w * e1y * e2x; S[1][1] += w * e1y * e2y; S[1][2] += w * e1y * e2z;
    S[2][0] += w * e1z * e2x; S[2][1] += w * e1z * e2y; S[2][2] += w * e1z * e2z;
  }
  S[0][0] += 1e-8f; S[1][1] += 1e-8f; S[2][2] += 1e-8f;

  // --- Proper rotation via eigen-decomposition of S^T S = V diag(sigma^2) V^T.
  float A[3][3];
#pragma unroll
  for (int p = 0; p < 3; ++p)
#pragma unroll
    for (int q = 0; q < 3; ++q)
      A[p][q] = S[0][p] * S[0][q] + S[1][p] * S[1][q] + S[2][p] * S[2][q];

  float V[3][3] = {{1.0f, 0.0f, 0.0f}, {0.0f, 1.0f, 0.0f}, {0.0f, 0.0f, 1.0f}};
#pragma unroll
  for (int sweep = 0; sweep < 4; ++sweep) {
    jrot<0, 1, 2>(A, V);
    jrot<0, 2, 1>(A, V);
    jrot<1, 2, 0>(A, V);
  }

  float l0 = A[0][0], l1 = A[1][1], l2 = A[2][2];
  // Sort eigenpairs descending (swap V columns along).
  if (l1 > l0) { float t = l0; l0 = l1; l1 = t;
    t = V[0][0]; V[0][0] = V[0][1]; V[0][1] = t;
    t = V[1][0]; V[1][0] = V[1][1]; V[1][1] = t;
    t = V[2][0]; V[2][0] = V[2][1]; V[2][1] = t; }
  if (l2 > l0) { float t = l0; l0 = l2; l2 = t;
    t = V[0][0]; V[0][0] = V[0][2]; V[0][2] = t;
    t = V[1][0]; V[1][0] = V[1][2]; V[1][2] = t;
    t = V[2][0]; V[2][0] = V[2][2]; V[2][2] = t; }
  if (l2 > l1) { float t = l1; l1 = l2; l2 = t;
    t = V[0][1]; V[0][1] = V[0][2]; V[0][2] = t;
    t = V[1][1]; V[1][1] = V[1][2]; V[1][2] = t;
    t = V[2][1]; V[2][1] = V[2][2]; V[2][2] = t; }

  const float det =
      S[0][0] * (S[1][1] * S[2][2] - S[1][2] * S[2][1]) -
      S[0][1] * (S[1][0] * S[2][2] - S[1][2] * S[2][0]) +
      S[0][2] * (S[1][0] * S[2][1] - S[1][1] * S[2][0]);

  float d0 = rsqrtf(fmaxf(l0, 1e-24f));   // 1/sigma_0
  float d1 = rsqrtf(fmaxf(l1, 1e-24f));
  float d2 = rsqrtf(fmaxf(l2, 1e-24f));
  if (det < 0.0f) d2 = -d2;               // flip smallest singular direction

  // R = V * diag(d) * V^T * S^T  ==  V * diag(1,1,sign(det)) * U^T
  float M[3][3];
#pragma unroll
  for (int a = 0; a < 3; ++a)
#pragma unroll
    for (int m = 0; m < 3; ++m)
      M[a][m] = d0 * V[a][0] * V[m][0] + d1 * V[a][1] * V[m][1] + d2 * V[a][2] * V[m][2];
  float R[3][3];
#pragma unroll
  for (int a = 0; a < 3; ++a)
#pragma unroll
    for (int c = 0; c < 3; ++c)
      R[a][c] = M[a][0] * S[c][0] + M[a][1] * S[c][1] + M[a][2] * S[c][2];

  // --- Pass 2: residual energy per axis (re-gather edges from LDS).
  float E0 = 0.0f, E1 = 0.0f, E2 = 0.0f;
#pragma unroll
  for (int k = 0; k < K_; ++k) {
    const int j = jv[k];
    const float w = wv[k];
    const float e1x = cx1 - s1[3 * j + 0];
    const float e1y = cy1 - s1[3 * j + 1];
    const float e1z = cz1 - s1[3 * j + 2];
    const float e2x = cx2 - s2[3 * j + 0];
    const float e2y = cy2 - s2[3 * j + 1];
    const float e2z = cz2 - s2[3 * j + 2];
    const float rx = e2x - (R[0][0] * e1x + R[0][1] * e1y + R[0][2] * e1z);
    const float ry = e2y - (R[1][0] * e1x + R[1][1] * e1y + R[1][2] * e1z);
    const float rz = e2z - (R[2][0] * e1x + R[2][1] * e1y + R[2][2] * e1z);
    E0 += w * rx * rx;
    E1 += w * ry * ry;
    E2 += w * rz * rz;
  }

  float* o = out + bn * 3;
  o[0] = 1e-4f * E0;
  o[1] = 1e-4f * E1;
  o[2] = 1e-4f * E2;
}

extern "C" void kernel_launch(void* const* d_in, const int* in_sizes, int n_in,
                              void* d_out, int out_size, void* d_ws, size_t ws_size,
                              hipStream_t stream) {
  (void)in_sizes; (void)n_in; (void)out_size; (void)d_ws; (void)ws_size;
  const float* xyz1 = (const float*)d_in[0];
  const float* xyz2 = (const float*)d_in[1];
  const int*   nl   = (const int*)d_in[2];
  const int*   nn   = (const int*)d_in[3];
  const float* wm   = (const float*)d_in[4];
  float* out = (float*)d_out;
  const size_t shmem = (size_t)6 * N_ * sizeof(float);  // 192 KB LDS staging
  arap_kernel<<<dim3(B_ * TILES_PER_BATCH), dim3(TPB), shmem, stream>>>(
      xyz1, xyz2, nl, nn, wm, out);
}